// binadamard_parametrization_79551384256559
// MI455X (gfx1250) — compile-verified
//
#include <hip/hip_runtime.h>

// out[i,j] = sign(sign_vector[i]) * H[i,j], N = 8192 (fp32).
// Pure streaming problem: 512 MB of traffic, ~22 us floor at 23.3 TB/s.
// Strategy: b128 loads/stores with gfx1250 non-temporal cache hints
// (dataset > 192 MB L2, zero reuse), uniform s_load of the per-row sign,
// 8 loads in flight per thread.

#define N_DIM 8192
#define BLOCK_THREADS 256
#define VEC4_PER_ROW (N_DIM / 4)                        // 2048 float4 per row
#define VEC4_PER_THREAD (VEC4_PER_ROW / BLOCK_THREADS)  // 8

// Native clang vector type: required by __builtin_nontemporal_load/store
// (HIP's float4 is a class and is rejected).
typedef float v4f __attribute__((ext_vector_type(4)));

__global__ __launch_bounds__(BLOCK_THREADS) void sign_row_scale_kernel(
    const float* __restrict__ sv,   // [N] sign vector
    const float* __restrict__ H,    // [N, N] block-Hadamard kernel
    float* __restrict__ out)        // [N, N]
{
    const int row = blockIdx.x;

    // Uniform across the block -> compiler emits a scalar (SMEM) load.
    const float v = sv[row];
    // jnp.sign semantics: +1 / -1 / 0
    const float s = (float)((v > 0.0f) - (v < 0.0f));

    const v4f* __restrict__ hrow =
        reinterpret_cast<const v4f*>(H + (size_t)row * N_DIM);
    v4f* __restrict__ orow =
        reinterpret_cast<v4f*>(out + (size_t)row * N_DIM);

    // 2048 float4 per row over 256 threads -> 8 b128 transactions per thread.
    // Full unroll lets all 8 global_load_b128 issue before the first
    // s_wait_loadcnt (deep LOADcnt pipelining).
#pragma unroll
    for (int k = 0; k < VEC4_PER_THREAD; ++k) {
        const int idx = threadIdx.x + k * BLOCK_THREADS;
        v4f h = __builtin_nontemporal_load(&hrow[idx]);
        v4f o = s * h;  // element-wise on native vector type
        __builtin_nontemporal_store(o, &orow[idx]);
    }
}

extern "C" void kernel_launch(void* const* d_in, const int* in_sizes, int n_in,
                              void* d_out, int out_size, void* d_ws, size_t ws_size,
                              hipStream_t stream) {
    // setup_inputs() order: [0]=weight (ignored), [1]=sign_vector, [2]=block_hadamard_kernel
    const float* sv = (const float*)d_in[1];
    const float* H  = (const float*)d_in[2];
    float* out      = (float*)d_out;

    const int rows = in_sizes[1];  // == N_DIM == 8192 per reference

    dim3 grid(rows);
    dim3 block(BLOCK_THREADS);
    hipLaunchKernelGGL(sign_row_scale_kernel, grid, block, 0, stream,
                       sv, H, out);
}